// ProteinEncoder_10814727651348
// MI455X (gfx1250) — compile-verified
//
#include <hip/hip_runtime.h>
#include <math.h>

// ---------------------------------------------------------------------------
// Problem constants (fixed by setup_inputs)
// ---------------------------------------------------------------------------
#define NBATCH          8
#define NNODES          4096
#define KNN_K           16
#define NBASIS          8
#define MUL1E           32
#define FEAT_OUT        464     // 64 zeros | 96 agg | 80 zeros | 112 atom | 112 res
#define DIM0            64
#define EMB_DIM         112

// kernel 1 tiling
#define ROWS_PER_WAVE   16
#define WAVES_PER_BLOCK 8
#define ROWS_PER_BLOCK  (ROWS_PER_WAVE * WAVES_PER_BLOCK)   // 128
#define CHUNK           64      // columns staged per iteration
#define RS              66      // padded score-row stride: conflict-free banks
                                //  scan  bank = (2*sub + 32*hi + q)        % 64
                                //  store bank = (2*v + 16*hi + 16*t + sub) % 64

// kernel 2 tiling
#define NODES_PER_BLOCK 16

typedef __attribute__((ext_vector_type(2))) float v2f;
typedef __attribute__((ext_vector_type(8))) float v8f;

// ---------------------------------------------------------------------------
// Kernel 1: WMMA-tiled KNN.
// Score(i,j) = |x_j|^2 - 2 x_i . x_j   (row-constant |x_i|^2 dropped: cannot
// change the per-row top-k ordering).  One V_WMMA_F32_16X16X4_F32 per 16x16
// tile:
//   A[16x4] row i = [xi, yi, zi, 1]          (lanes 0-15: K0,K1; 16-31: K2,K3)
//   B[4x16] col j = [-2xj, -2yj, -2zj, nj]   (VGPR0: K0/K2, VGPR1: K1/K3)
// Each wave owns 16 rows; 2 lanes per row keep private sorted top-16 lists in
// VGPRs; self column poisoned in LDS; scan is b64 loads + min4-guarded insert.
// ---------------------------------------------------------------------------
__global__ __launch_bounds__(256) void knn_kernel(const float* __restrict__ coords,
                                                  int* __restrict__ knn)
{
    const int lane = threadIdx.x & 31;
    const int wave = threadIdx.x >> 5;
    const int sub  = lane & 15;      // 0..15  (row / col within tile)
    const int hi   = lane >> 4;      // 0 or 1 (K-halves of WMMA operands)

    const int batch    = blockIdx.x;
    const int row_base = blockIdx.y * ROWS_PER_BLOCK + wave * ROWS_PER_WAVE;

    __shared__ float s_col[2 * CHUNK * 4];                         // double-buffered
    __shared__ float s_score[WAVES_PER_BLOCK * ROWS_PER_WAVE * RS];
    float* my_score = &s_score[wave * ROWS_PER_WAVE * RS];

    const float* cb = coords + (size_t)batch * NNODES * 3;

    // stage CHUNK columns pre-scaled: (-2x, -2y, -2z, |x|^2)
    auto stage = [&](int c0, int buf) {
        if (threadIdx.x < CHUNK) {
            const int col = c0 + threadIdx.x;
            const float x = cb[col * 3 + 0];
            const float y = cb[col * 3 + 1];
            const float z = cb[col * 3 + 2];
            float* d = &s_col[(buf * CHUNK + threadIdx.x) * 4];
            d[0] = -2.0f * x;
            d[1] = -2.0f * y;
            d[2] = -2.0f * z;
            d[3] = x * x + y * y + z * z;
        }
    };

    // ---- A operand: fixed for the whole kernel (this wave's 16 rows) ----
    const int arow = row_base + sub;
    const float ax = cb[arow * 3 + 0];
    const float ay = cb[arow * 3 + 1];
    const float az = cb[arow * 3 + 2];
    v2f A;
    A.x = hi ? az   : ax;    // lanes 0-15 -> K=0 ; lanes 16-31 -> K=2
    A.y = hi ? 1.0f : ay;    // lanes 0-15 -> K=1 ; lanes 16-31 -> K=3

    // ---- register-resident sorted top-16 (ascending; ks[15] = current worst) ----
    float ks[KNN_K];
    int   ki[KNN_K];
#pragma unroll
    for (int p = 0; p < KNN_K; ++p) { ks[p] = 3.0e38f; ki[p] = 0; }

    // cmov insertion into the sorted list (keeps ks/ki in VGPRs)
    auto insert = [&](float s, int col) {
        if (s < ks[KNN_K - 1]) {
#pragma unroll
            for (int p = KNN_K - 1; p >= 1; --p) {
                const float a  = ks[p - 1];
                const int   ai = ki[p - 1];
                const bool shift = (a > s);
                const float kp = ks[p];
                const int  kip = ki[p];
                const bool place = (!shift) && (kp > s);
                ks[p] = shift ? a  : (place ? s   : kp);
                ki[p] = shift ? ai : (place ? col : kip);
            }
            if (ks[0] > s) { ks[0] = s; ki[0] = col; }
        }
    };

    const int myrow_g = row_base + sub;

    // scan base: invariant across chunks; 8B-aligned for every lane
    // (sub*264B + hi*128B), banks pairwise disjoint for b64 loads.
    const float* scanp = &my_score[sub * RS + hi * (CHUNK / 2)];
    float* poisonp = &my_score[sub * RS];

    stage(0, 0);
    __syncthreads();

    const int NCHUNK = NNODES / CHUNK;
    for (int ci = 0; ci < NCHUNK; ++ci) {
        const int c0 = ci * CHUNK;
        if (ci + 1 < NCHUNK) stage(c0 + CHUNK, (ci + 1) & 1);   // prefetch next
        const float* colb = &s_col[(ci & 1) * CHUNK * 4];

        // 4 column sub-tiles of 16 -> 4 WMMAs, D tiles into padded LDS scratch
#pragma unroll
        for (int t = 0; t < CHUNK / 16; ++t) {
            const int cc = t * 16 + sub;
            const float m2x = colb[cc * 4 + 0];
            const float m2y = colb[cc * 4 + 1];
            const float m2z = colb[cc * 4 + 2];
            const float n2  = colb[cc * 4 + 3];
            v2f B;
            B.x = hi ? m2z : m2x;    // K=2 / K=0
            B.y = hi ? n2  : m2y;    // K=3 / K=1
            v8f C = {};
            C = __builtin_amdgcn_wmma_f32_16x16x4_f32(
                    /*neg_a=*/false, A, /*neg_b=*/false, B,
                    /*c_mod=*/(short)0, C, /*reuse_a=*/false, /*reuse_b=*/false);
            // D layout: VGPR v, lanes 0-15 -> row v, lanes 16-31 -> row v+8
#pragma unroll
            for (int v = 0; v < 8; ++v)
                my_score[(v + 8 * hi) * RS + t * 16 + sub] = C[v];
        }

        // poison the diagonal (self edge) so the scan needs no index compare
        {
            const int dcol = myrow_g - c0;
            if (dcol >= 0 && dcol < CHUNK && ((dcol >> 5) == hi))
                poisonp[dcol] = 3.0e38f;
        }

        // scan: b64 pairs, one min4 guard per 4 candidates
        const int colbase = c0 + hi * (CHUNK / 2);
#pragma unroll 2
        for (int q = 0; q < CHUNK / 2; q += 4) {
            const float2 ab = *(const float2*)(scanp + q);
            const float2 cd = *(const float2*)(scanp + q + 2);
            const float m4 = fminf(fminf(ab.x, ab.y), fminf(cd.x, cd.y));
            if (m4 < ks[KNN_K - 1]) {
                insert(ab.x, colbase + q + 0);
                insert(ab.y, colbase + q + 1);
                insert(cd.x, colbase + q + 2);
                insert(cd.y, colbase + q + 3);
            }
        }
        __syncthreads();
    }

    // ---- merge the two half-row lists; reuse score scratch ----
#pragma unroll
    for (int p = 0; p < KNN_K; ++p) {
        my_score[sub * RS + hi * KNN_K + p]      = ks[p];
        my_score[sub * RS + 32 + hi * KNN_K + p] = __int_as_float(ki[p]);
    }
    __syncthreads();

    if (hi == 0) {
        int i0 = 0, i1 = 0;
        const int out_base = (batch * NNODES + row_base + sub) * KNN_K;
#pragma unroll
        for (int k = 0; k < KNN_K; ++k) {
            const float s0 = (i0 < KNN_K) ? my_score[sub * RS + i0]         : 3.0e38f;
            const float s1 = (i1 < KNN_K) ? my_score[sub * RS + KNN_K + i1] : 3.0e38f;
            const bool take0 = (s0 <= s1);
            const int idx = take0
                ? __float_as_int(my_score[sub * RS + 32 + i0])
                : __float_as_int(my_score[sub * RS + 32 + KNN_K + i1]);
            knn[out_base + k] = batch * NNODES + idx;   // flat sender index
            i0 += take0 ? 1 : 0;
            i1 += take0 ? 0 : 1;
        }
    }
}

// ---------------------------------------------------------------------------
// Kernel 2: fused radial-basis -> tensor product -> mean -> concat output.
// Exactly K=16 edges per receiver, so scatter_mean == average over own list.
// ---------------------------------------------------------------------------
__global__ __launch_bounds__(256) void feat_kernel(
    const float* __restrict__ coords,     // [B*N,3]
    const int*   __restrict__ atom_types, // [B*N]
    const int*   __restrict__ res_types,  // [B*N]
    const float* __restrict__ tp_weight,  // [8,32]
    const float* __restrict__ atom_emb,   // [20,112]
    const float* __restrict__ res_emb,    // [21,112]
    const int*   __restrict__ knn,        // [B*N,16] flat sender idx
    float*       __restrict__ out)        // [B*N,464]
{
    __shared__ float s_w[NBASIS * MUL1E];                       // 256
    __shared__ float s_rb[NODES_PER_BLOCK * KNN_K * NBASIS];    // 2048
    __shared__ float s_rhat[NODES_PER_BLOCK * KNN_K * 3];       // 768

    const int t     = threadIdx.x;
    const int node0 = blockIdx.x * NODES_PER_BLOCK;

    s_w[t] = tp_weight[t];   // 256 threads, 256 weights

    // ---- phase A: per-edge geometry (1 thread per edge) ----
    {
        const int nl = t >> 4, j = t & 15;
        const int node = node0 + nl;
        const int nbr  = knn[node * KNN_K + j];
        const float rx = coords[nbr * 3 + 0] - coords[node * 3 + 0];
        const float ry = coords[nbr * 3 + 1] - coords[node * 3 + 1];
        const float rz = coords[nbr * 3 + 2] - coords[node * 3 + 2];
        const float dist = sqrtf(rx * rx + ry * ry + rz * rz);
        const float inv  = 1.0f / (dist + 1e-8f);
        float u = dist * 0.1f;                       // dist / MAX_RADIUS
        u = fminf(fmaxf(u, 0.0f), 1.0f);
        float g[NBASIS];
        float sum = 0.0f;
#pragma unroll
        for (int v = 0; v < NBASIS; ++v) {           // centers v/7, sigma=1/8 -> 1/(2s^2)=32
            const float d = u - (float)v * (1.0f / 7.0f);
            g[v] = __expf(-d * d * 32.0f);
            sum += g[v];
        }
        const float rnorm = 1.0f / sum;              // sum of exps > 0 always
        const int e = nl * KNN_K + j;
#pragma unroll
        for (int v = 0; v < NBASIS; ++v) s_rb[e * NBASIS + v] = g[v] * rnorm;
        s_rhat[e * 3 + 0] = rx * inv;
        s_rhat[e * 3 + 1] = ry * inv;
        s_rhat[e * 3 + 2] = rz * inv;
    }
    __syncthreads();

    // ---- phase B: agg_1e[node][m][d]; thread owns (node_local, m) pairs ----
    const float inv_scale = 1.0f / (16.0f * 1.7320508075688772f);   // mean * 1/sqrt(3)
#pragma unroll
    for (int it = 0; it < (NODES_PER_BLOCK * MUL1E) / 256; ++it) {  // 2 iters
        const int cg = t + it * 256;
        const int nl = cg >> 5;
        const int m  = cg & 31;
        float a0 = 0.0f, a1 = 0.0f, a2 = 0.0f;
#pragma unroll 4
        for (int j = 0; j < KNN_K; ++j) {
            const int e = nl * KNN_K + j;
            float w = 0.0f;
#pragma unroll
            for (int v = 0; v < NBASIS; ++v) w += s_rb[e * NBASIS + v] * s_w[v * MUL1E + m];
            a0 += w * s_rhat[e * 3 + 0];
            a1 += w * s_rhat[e * 3 + 1];
            a2 += w * s_rhat[e * 3 + 2];
        }
        const size_t ob = (size_t)(node0 + nl) * FEAT_OUT + DIM0 + m * 3;
        out[ob + 0] = a0 * inv_scale;
        out[ob + 1] = a1 * inv_scale;
        out[ob + 2] = a2 * inv_scale;
    }

    // ---- phase C: zeros (64 + 80) and embedding gathers (112 + 112) ----
    for (int idx = t; idx < NODES_PER_BLOCK * 368; idx += 256) {
        const int nl = idx / 368;
        const int c2 = idx % 368;
        const int node = node0 + nl;
        const size_t ob = (size_t)node * FEAT_OUT;
        float val;
        int off;
        if (c2 < 64)        { off = c2;              val = 0.0f; }
        else if (c2 < 144)  { off = 160 + (c2 - 64); val = 0.0f; }
        else if (c2 < 256)  { off = 240 + (c2 - 144);
                              val = atom_emb[atom_types[node] * EMB_DIM + (c2 - 144)]; }
        else                { off = 352 + (c2 - 256);
                              val = res_emb[res_types[node] * EMB_DIM + (c2 - 256)]; }
        out[ob + off] = val;
    }
}

// ---------------------------------------------------------------------------
extern "C" void kernel_launch(void* const* d_in, const int* in_sizes, int n_in,
                              void* d_out, int out_size, void* d_ws, size_t ws_size,
                              hipStream_t stream)
{
    const float* coords     = (const float*)d_in[0];   // [8,4096,3]
    const int*   atom_types = (const int*)  d_in[1];   // [8,4096]
    const int*   res_types  = (const int*)  d_in[2];   // [8,4096]
    const float* tp_weight  = (const float*)d_in[3];   // [8,32]
    const float* atom_emb   = (const float*)d_in[4];   // [20,112]
    const float* res_emb    = (const float*)d_in[5];   // [21,112]
    float* out = (float*)d_out;                        // [8,4096,464]
    int*   knn = (int*)d_ws;                           // [8*4096,16] scratch

    dim3 g1(NBATCH, NNODES / ROWS_PER_BLOCK);          // (8, 32)
    knn_kernel<<<g1, 256, 0, stream>>>(coords, knn);

    const int g2 = (NBATCH * NNODES) / NODES_PER_BLOCK; // 2048
    feat_kernel<<<g2, 256, 0, stream>>>(coords, atom_types, res_types,
                                        tp_weight, atom_emb, res_emb, knn, out);
}